// PretrainEncoder_74388833566993
// MI455X (gfx1250) — compile-verified
//
#include <hip/hip_runtime.h>
#include <math.h>

namespace {

constexpr int N_NODES = 100000;
constexpr int N_EDGES = 1600000;
constexpr int N_HEAVY = 25000;
constexpr int MSG_S = 32;
constexpr int MSG_V = 16;
constexpr int XDIM  = 64;   // padded node feature stride
constexpr int MSGD  = 80;   // message width

typedef __attribute__((ext_vector_type(2))) float v2f;
typedef __attribute__((ext_vector_type(8))) float v8f;

__device__ __forceinline__ v8f wmma4(v2f a, v2f b, v8f c) {
  // D(16x16 f32) = A(16x4 f32) * B(4x16 f32) + C
  return __builtin_amdgcn_wmma_f32_16x16x4_f32(false, a, false, b, (short)0, c,
                                               false, false);
}

// wave-private LDS producer->consumer handoff: HW keeps LDS in-order per wave,
// the fence stops the compiler from reordering and drains DScnt.
__device__ __forceinline__ void lds_fence() {
  asm volatile("s_wait_dscnt 0x0" ::: "memory");
}

// ---------------------------------------------------------------------------
// x[n, 0:48] = embed_w[node_idx[n]]
__global__ void embed_kernel(const int* __restrict__ node_idx,
                             const float* __restrict__ embed_w,
                             float* __restrict__ x) {
  int t = blockIdx.x * 256 + threadIdx.x;
  if (t >= N_NODES * 48) return;
  int n = t / 48, c = t % 48;
  x[(size_t)n * XDIM + c] = embed_w[(size_t)node_idx[n] * 48 + c];
}

// ---------------------------------------------------------------------------
// Edge tensor product -> scatter-add into node_msg[N,80].
// Wave handles 32 edges = two 16-edge WMMA sub-tiles.
//   scalar out : a[KA=CS+CV]    @ Wss[KA,32]
//   vector out : b_d[KB=CS+2CV] @ Wv [KB,16]   (d = x,y,z; Wv shared across d)
template <int CS, int CV>
__global__ __launch_bounds__(128) void edge_tp_kernel(
    const float* __restrict__ x, const int* __restrict__ src,
    const int* __restrict__ dst, const float* __restrict__ ea,
    const float* __restrict__ w_ss, const float* __restrict__ w_vv,
    const float* __restrict__ w_sv, const float* __restrict__ w_vs,
    const float* __restrict__ w_xx, float* __restrict__ node_msg) {
  constexpr int KA = CS + CV;
  constexpr int KB = CS + 2 * CV;
  constexpr int KA_CH = KA / 4;
  constexpr int KB_CH = KB / 4;
  constexpr int NW = 4;
  static_assert(KA % 4 == 0 && KB % 4 == 0, "K must be multiple of 4");

  __shared__ __align__(16) v2f lws[KA_CH * 2][32];  // scalar-out B operands
  __shared__ __align__(16) v2f lwv[KB_CH][32];      // vector-out B operands
  __shared__ __align__(16) float la[NW][32][KA];    // per-wave A features
  __shared__ __align__(16) float lb[NW][32][KB];

  const int tid = threadIdx.x;
  const int l = tid & 31, wv = tid >> 5;
  const int h = l >> 4, q = l & 15;

  // Stage B operands in WMMA lane layout: B.x = W[4c+2h][o], B.y = W[4c+2h+1][o]
  for (int c = wv; c < KA_CH; c += NW) {
    int k0 = 4 * c + 2 * h;
    for (int t = 0; t < 2; ++t) {
      int o = 16 * t + q;
      float b0 = (k0 < CS) ? w_ss[k0 * MSG_S + o] : w_vv[(k0 - CS) * MSG_S + o];
      float b1 = (k0 + 1 < CS) ? w_ss[(k0 + 1) * MSG_S + o]
                               : w_vv[(k0 + 1 - CS) * MSG_S + o];
      v2f wv2 = {b0, b1};
      lws[c * 2 + t][l] = wv2;
    }
  }
  for (int c = wv; c < KB_CH; c += NW) {
    int k0 = 4 * c + 2 * h;
    float bb[2];
    for (int u = 0; u < 2; ++u) {
      int k = k0 + u;
      float w;
      if (k < CS)            w = w_sv[k * MSG_V + q];
      else if (k < CS + CV)  w = w_vs[(k - CS) * MSG_V + q];
      else                   w = w_xx[(k - CS - CV) * MSG_V + q];
      bb[u] = w;
    }
    v2f wv2 = {bb[0], bb[1]};
    lwv[c][l] = wv2;
  }
  __syncthreads();

  const int ntiles = (N_EDGES + 31) / 32;
  for (int tile = blockIdx.x * NW + wv; tile < ntiles; tile += gridDim.x * NW) {
    const int e = tile * 32 + l;
    const bool valid = e < N_EDGES;
    float es = 0.f, evx = 0.f, evy = 0.f, evz = 0.f;
    float xs[CS], xvx[CV], xvy[CV], xvz[CV];
#pragma unroll
    for (int i = 0; i < CS; ++i) xs[i] = 0.f;
#pragma unroll
    for (int j = 0; j < CV; ++j) { xvx[j] = 0.f; xvy[j] = 0.f; xvz[j] = 0.f; }
    if (valid) {
      const float* ep = ea + (size_t)e * 4;
      es = ep[0]; evx = ep[1]; evy = ep[2]; evz = ep[3];
      const float* xp = x + (size_t)src[e] * XDIM;
#pragma unroll
      for (int i = 0; i < CS; ++i) xs[i] = xp[i];
#pragma unroll
      for (int j = 0; j < CV; ++j) {
        xvx[j] = xp[CS + 3 * j + 0];
        xvy[j] = xp[CS + 3 * j + 1];
        xvz[j] = xp[CS + 3 * j + 2];
      }
    }

    // ---- scalar-path features: a[i]=xs_i*es ; a[CS+j]=xv_j . ev
    {
      float f[KA];
#pragma unroll
      for (int i = 0; i < CS; ++i) f[i] = xs[i] * es;
#pragma unroll
      for (int j = 0; j < CV; ++j)
        f[CS + j] = xvx[j] * evx + xvy[j] * evy + xvz[j] * evz;
#pragma unroll
      for (int k = 0; k < KA; k += 4)
        *(float4*)&la[wv][l][k] = make_float4(f[k], f[k + 1], f[k + 2], f[k + 3]);
    }
    lds_fence();
#pragma unroll
    for (int sub = 0; sub < 2; ++sub) {
      v8f acc0 = {}, acc1 = {};
#pragma unroll
      for (int c = 0; c < KA_CH; ++c) {
        v2f A = *(const v2f*)&la[wv][sub * 16 + q][4 * c + 2 * h];
        acc0 = wmma4(A, lws[c * 2 + 0][l], acc0);
        acc1 = wmma4(A, lws[c * 2 + 1][l], acc1);
      }
#pragma unroll
      for (int r = 0; r < 8; ++r) {
        int eg = tile * 32 + sub * 16 + r + 8 * h;
        if (eg < N_EDGES) {
          float* np = node_msg + (size_t)dst[eg] * MSGD;
          atomicAdd(np + q, acc0[r]);
          atomicAdd(np + 16 + q, acc1[r]);
        }
      }
    }
    lds_fence();

    // ---- vector-path features per dimension d
#pragma unroll
    for (int d = 0; d < 3; ++d) {
      const float evd = (d == 0) ? evx : ((d == 1) ? evy : evz);
      {
        float f[KB];
#pragma unroll
        for (int i = 0; i < CS; ++i) f[i] = xs[i] * evd;
#pragma unroll
        for (int j = 0; j < CV; ++j) {
          float xvd = (d == 0) ? xvx[j] : ((d == 1) ? xvy[j] : xvz[j]);
          f[CS + j] = xvd * es;
          float cr;
          if (d == 0)      cr = xvy[j] * evz - xvz[j] * evy;
          else if (d == 1) cr = xvz[j] * evx - xvx[j] * evz;
          else             cr = xvx[j] * evy - xvy[j] * evx;
          f[CS + CV + j] = cr;
        }
#pragma unroll
        for (int k = 0; k < KB; k += 4)
          *(float4*)&lb[wv][l][k] =
              make_float4(f[k], f[k + 1], f[k + 2], f[k + 3]);
      }
      lds_fence();
#pragma unroll
      for (int sub = 0; sub < 2; ++sub) {
        v8f acc = {};
#pragma unroll
        for (int c = 0; c < KB_CH; ++c) {
          v2f A = *(const v2f*)&lb[wv][sub * 16 + q][4 * c + 2 * h];
          acc = wmma4(A, lwv[c][l], acc);
        }
#pragma unroll
        for (int r = 0; r < 8; ++r) {
          int eg = tile * 32 + sub * 16 + r + 8 * h;
          if (eg < N_EDGES) {
            float* np = node_msg + (size_t)dst[eg] * MSGD;
            atomicAdd(np + MSG_S + q * 3 + d, acc[r]);
          }
        }
      }
      lds_fence();
    }
  }
}

// ---------------------------------------------------------------------------
// Gate: x_aggr from node_msg; also heavy scatter-sum into haggr/hcnt.
__global__ __launch_bounds__(128) void gate_kernel(
    const float* __restrict__ node_msg, const float* __restrict__ lin_s,
    const float* __restrict__ lin_v, const int* __restrict__ z,
    const int* __restrict__ canonical, float* __restrict__ x,
    float* __restrict__ haggr, float* __restrict__ hcnt) {
  __shared__ float ls[32 * 32];
  __shared__ float lv[16 * 16];
  for (int i = threadIdx.x; i < 1024; i += blockDim.x) ls[i] = lin_s[i];
  for (int i = threadIdx.x; i < 256; i += blockDim.x) lv[i] = lin_v[i];
  __syncthreads();
  int n = blockIdx.x * blockDim.x + threadIdx.x;
  if (n >= N_NODES) return;
  const float* m = node_msg + (size_t)n * MSGD;
  float ns[32], gs[32];
#pragma unroll
  for (int i = 0; i < 32; ++i) ns[i] = m[i];
#pragma unroll
  for (int o = 0; o < 32; ++o) gs[o] = 0.f;
#pragma unroll
  for (int k = 0; k < 32; ++k) {
    float a = ns[k];
#pragma unroll
    for (int o = 0; o < 32; ++o) gs[o] += a * ls[k * 32 + o];
  }
  float nvx[16], nvy[16], nvz[16];
#pragma unroll
  for (int v = 0; v < 16; ++v) {
    nvx[v] = m[32 + 3 * v + 0];
    nvy[v] = m[32 + 3 * v + 1];
    nvz[v] = m[32 + 3 * v + 2];
  }
  float out[64];
#pragma unroll
  for (int o = 0; o < 16; ++o) out[o] = 1.f / (1.f + expf(-gs[o]));
#pragma unroll
  for (int u = 0; u < 16; ++u) {
    float gx = 0.f, gy = 0.f, gz = 0.f;
#pragma unroll
    for (int v = 0; v < 16; ++v) {
      float w = lv[v * 16 + u];
      gx += nvx[v] * w; gy += nvy[v] * w; gz += nvz[v] * w;
    }
    float g = tanhf(gs[16 + u]);
    out[16 + 3 * u + 0] = gx * g;
    out[16 + 3 * u + 1] = gy * g;
    out[16 + 3 * u + 2] = gz * g;
  }
  float* xo = x + (size_t)n * XDIM;
#pragma unroll
  for (int c = 0; c < 64; ++c) xo[c] = out[c];
  if (z[n] > 1) {
    float* hb = haggr + (size_t)canonical[n] * XDIM;
#pragma unroll
    for (int c = 0; c < 64; ++c) atomicAdd(hb + c, out[c]);
    atomicAdd(hcnt + canonical[n], 1.0f);
  }
}

// ---------------------------------------------------------------------------
// Heavy self-TP, scalar outputs: so = f[512] @ Ws[512,16]
//   k in [0,256): hs_i*hs_j ; k in [256,512): hv_i . hv_j
// j-side values [8h, 8h+8) are loaded with the runtime base folded into the
// ADDRESS (not a register-array index) to avoid cndmask select chains.
__global__ __launch_bounds__(128) void heavy_tp_s_kernel(
    const float* __restrict__ haggr, const float* __restrict__ hcnt,
    const float* __restrict__ w_ss, const float* __restrict__ w_vv,
    float* __restrict__ xtp) {
  constexpr int NW = 4;
  __shared__ __align__(16) v2f lw[128][32];        // 32 KB
  __shared__ __align__(16) float la[NW][16][64];   // 16 KB

  const int tid = threadIdx.x;
  const int l = tid & 31, wv = tid >> 5;
  const int h = l >> 4, q = l & 15;

  for (int c = wv; c < 128; c += NW) {
    int k0 = 4 * c + 2 * h;
    float b0 = (k0 < 256) ? w_ss[k0 * 16 + q] : w_vv[(k0 - 256) * 16 + q];
    float b1 = (k0 + 1 < 256) ? w_ss[(k0 + 1) * 16 + q]
                              : w_vv[(k0 + 1 - 256) * 16 + q];
    v2f wv2 = {b0, b1};
    lw[c][l] = wv2;
  }
  __syncthreads();

  const int ntiles = (N_HEAVY + 15) / 16;
  for (int tile = blockIdx.x * NW + wv; tile < ntiles; tile += gridDim.x * NW) {
    const int a = tile * 16 + q;
    float hs[16], hvx[16], hvy[16], hvz[16];       // i-side (const indexed)
    float hsj[8], hvxj[8], hvyj[8], hvzj[8];       // j-side, base 8h in addr
#pragma unroll
    for (int i = 0; i < 16; ++i) { hs[i] = 0.f; hvx[i] = 0.f; hvy[i] = 0.f; hvz[i] = 0.f; }
#pragma unroll
    for (int j = 0; j < 8; ++j) { hsj[j] = 0.f; hvxj[j] = 0.f; hvyj[j] = 0.f; hvzj[j] = 0.f; }
    if (a < N_HEAVY) {
      float rc = 1.f / fmaxf(hcnt[a], 1.f);
      const float* hp = haggr + (size_t)a * XDIM;
#pragma unroll
      for (int i = 0; i < 16; ++i) hs[i] = hp[i] * rc;
#pragma unroll
      for (int j = 0; j < 16; ++j) {
        hvx[j] = hp[16 + 3 * j + 0] * rc;
        hvy[j] = hp[16 + 3 * j + 1] * rc;
        hvz[j] = hp[16 + 3 * j + 2] * rc;
      }
      const int jb = 8 * h;
#pragma unroll
      for (int j = 0; j < 8; ++j) {
        hsj[j]  = hp[jb + j] * rc;
        hvxj[j] = hp[16 + 3 * (jb + j) + 0] * rc;
        hvyj[j] = hp[16 + 3 * (jb + j) + 1] * rc;
        hvzj[j] = hp[16 + 3 * (jb + j) + 2] * rc;
      }
    }
    v8f acc = {};
#pragma unroll
    for (int blk = 0; blk < 8; ++blk) {  // 0..3: ss, 4..7: vv
#pragma unroll
      for (int ii = 0; ii < 4; ++ii) {
        int i = (blk & 3) * 4 + ii;
        float f[8];
#pragma unroll
        for (int jj = 0; jj < 8; ++jj) {
          f[jj] = (blk < 4)
                      ? hs[i] * hsj[jj]
                      : hvx[i] * hvxj[jj] + hvy[i] * hvyj[jj] + hvz[i] * hvzj[jj];
        }
        *(float4*)&la[wv][q][ii * 16 + 8 * h] = make_float4(f[0], f[1], f[2], f[3]);
        *(float4*)&la[wv][q][ii * 16 + 8 * h + 4] = make_float4(f[4], f[5], f[6], f[7]);
      }
      lds_fence();
#pragma unroll
      for (int cc = 0; cc < 16; ++cc) {
        v2f A = *(const v2f*)&la[wv][q][cc * 4 + 2 * h];
        acc = wmma4(A, lw[blk * 16 + cc][l], acc);
      }
      lds_fence();
    }
#pragma unroll
    for (int r = 0; r < 8; ++r) {
      int ai = tile * 16 + r + 8 * h;
      if (ai < N_HEAVY) xtp[(size_t)ai * XDIM + q] = acc[r];
    }
  }
}

#define HVD(d, j)  ((d) == 0 ? hvx[j]  : ((d) == 1 ? hvy[j]  : hvz[j]))
#define HVDJ(d, j) ((d) == 0 ? hvxj[j] : ((d) == 1 ? hvyj[j] : hvzj[j]))

// Heavy self-TP, vector outputs per d: vo_d = f_d[768] @ Wv[768,16]
//   k<256: hs_i*hv_jd ; k<512: hv_id*hs_j ; k<768: cross(hv_i,hv_j)_d
__global__ __launch_bounds__(64) void heavy_tp_v_kernel(
    const float* __restrict__ haggr, const float* __restrict__ hcnt,
    const float* __restrict__ w_sv, const float* __restrict__ w_vs,
    const float* __restrict__ w_xx, float* __restrict__ xtp) {
  constexpr int NW = 2;
  __shared__ __align__(16) v2f lw[192][32];        // 48 KB
  __shared__ __align__(16) float la[NW][16][64];   //  8 KB

  const int tid = threadIdx.x;
  const int l = tid & 31, wv = tid >> 5;
  const int h = l >> 4, q = l & 15;

  for (int c = wv; c < 192; c += NW) {
    int k0 = 4 * c + 2 * h;
    float bb[2];
    for (int u = 0; u < 2; ++u) {
      int k = k0 + u;
      float w;
      if (k < 256)      w = w_sv[k * 16 + q];
      else if (k < 512) w = w_vs[(k - 256) * 16 + q];
      else              w = w_xx[(k - 512) * 16 + q];
      bb[u] = w;
    }
    v2f wv2 = {bb[0], bb[1]};
    lw[c][l] = wv2;
  }
  __syncthreads();

  const int ntiles = (N_HEAVY + 15) / 16;
  for (int tile = blockIdx.x * NW + wv; tile < ntiles; tile += gridDim.x * NW) {
    const int a = tile * 16 + q;
    float hs[16], hvx[16], hvy[16], hvz[16];
    float hsj[8], hvxj[8], hvyj[8], hvzj[8];
#pragma unroll
    for (int i = 0; i < 16; ++i) { hs[i] = 0.f; hvx[i] = 0.f; hvy[i] = 0.f; hvz[i] = 0.f; }
#pragma unroll
    for (int j = 0; j < 8; ++j) { hsj[j] = 0.f; hvxj[j] = 0.f; hvyj[j] = 0.f; hvzj[j] = 0.f; }
    if (a < N_HEAVY) {
      float rc = 1.f / fmaxf(hcnt[a], 1.f);
      const float* hp = haggr + (size_t)a * XDIM;
#pragma unroll
      for (int i = 0; i < 16; ++i) hs[i] = hp[i] * rc;
#pragma unroll
      for (int j = 0; j < 16; ++j) {
        hvx[j] = hp[16 + 3 * j + 0] * rc;
        hvy[j] = hp[16 + 3 * j + 1] * rc;
        hvz[j] = hp[16 + 3 * j + 2] * rc;
      }
      const int jb = 8 * h;
#pragma unroll
      for (int j = 0; j < 8; ++j) {
        hsj[j]  = hp[jb + j] * rc;
        hvxj[j] = hp[16 + 3 * (jb + j) + 0] * rc;
        hvyj[j] = hp[16 + 3 * (jb + j) + 1] * rc;
        hvzj[j] = hp[16 + 3 * (jb + j) + 2] * rc;
      }
    }
#pragma unroll
    for (int d = 0; d < 3; ++d) {
      v8f acc = {};
#pragma unroll
      for (int blk = 0; blk < 12; ++blk) {
        const int seg = blk >> 2;
#pragma unroll
        for (int ii = 0; ii < 4; ++ii) {
          int i = (blk & 3) * 4 + ii;
          float f[8];
#pragma unroll
          for (int jj = 0; jj < 8; ++jj) {
            float val;
            if (seg == 0)      val = hs[i] * HVDJ(d, jj);
            else if (seg == 1) val = HVD(d, i) * hsj[jj];
            else {
              if (d == 0)      val = hvy[i] * hvzj[jj] - hvz[i] * hvyj[jj];
              else if (d == 1) val = hvz[i] * hvxj[jj] - hvx[i] * hvzj[jj];
              else             val = hvx[i] * hvyj[jj] - hvy[i] * hvxj[jj];
            }
            f[jj] = val;
          }
          *(float4*)&la[wv][q][ii * 16 + 8 * h] = make_float4(f[0], f[1], f[2], f[3]);
          *(float4*)&la[wv][q][ii * 16 + 8 * h + 4] = make_float4(f[4], f[5], f[6], f[7]);
        }
        lds_fence();
#pragma unroll
        for (int cc = 0; cc < 16; ++cc) {
          v2f A = *(const v2f*)&la[wv][q][cc * 4 + 2 * h];
          acc = wmma4(A, lw[blk * 16 + cc][l], acc);
        }
        lds_fence();
      }
#pragma unroll
      for (int r = 0; r < 8; ++r) {
        int ai = tile * 16 + r + 8 * h;
        if (ai < N_HEAVY) xtp[(size_t)ai * XDIM + 16 + q * 3 + d] = acc[r];
      }
    }
  }
}
#undef HVD
#undef HVDJ

// ---------------------------------------------------------------------------
__global__ void merge_kernel(const float* __restrict__ xtp,
                             const int* __restrict__ z,
                             const int* __restrict__ canonical,
                             float* __restrict__ x) {
  int t = blockIdx.x * 256 + threadIdx.x;
  if (t >= N_NODES * XDIM) return;
  int n = t >> 6, c = t & 63;
  if (z[n] > 1) x[t] = xtp[(size_t)canonical[n] * XDIM + c];
}

__global__ void head_kernel(const float* __restrict__ x,
                            const float* __restrict__ hw,
                            const float* __restrict__ hb,
                            float* __restrict__ out) {
  int n = blockIdx.x * 256 + threadIdx.x;
  if (n >= N_NODES) return;
  const float* xp = x + (size_t)n * XDIM;
  float acc = hb[0];
#pragma unroll
  for (int c = 0; c < XDIM; ++c) acc += xp[c] * hw[c];
  out[n] = acc;
}

}  // namespace

extern "C" void kernel_launch(void* const* d_in, const int* in_sizes, int n_in,
                              void* d_out, int out_size, void* d_ws,
                              size_t ws_size, hipStream_t stream) {
  (void)n_in; (void)out_size; (void)ws_size;
  const int*   node_idx  = (const int*)d_in[0];
  const int*   src       = (const int*)d_in[1];
  const int*   dst       = src + N_EDGES;
  const float* edge_attr = (const float*)d_in[2];
  const int*   z         = (const int*)d_in[3];
  const int*   canonical = (const int*)d_in[4];
  const float* embed_w   = (const float*)d_in[5];

  // layer_params leaves at indices 6..29. Two possible flatten orders:
  //  insertion: tp_msg{ss,vv,sv,vs,xx}, lin_s, lin_v, tp_heavy{ss,vv,sv,vs,xx}
  //  pytree-sorted: lin_s, lin_v, tp_heavy{ss,sv,vs,vv,xx}, tp_msg{ss,sv,vs,vv,xx}
  // Distinguish via in_sizes[6]: 768 (=24*1*32 w_ss) vs 1024 (=32*32 lin_s).
  struct LayerP { const float* tm[5]; const float* lin_s; const float* lin_v; const float* th[5]; };
  LayerP L[2];
  const bool sorted_scheme = (in_sizes[6] == 1024);
  for (int li = 0; li < 2; ++li) {
    int b = 6 + 12 * li;
    if (!sorted_scheme) {
      for (int i = 0; i < 5; ++i) L[li].tm[i] = (const float*)d_in[b + i];
      L[li].lin_s = (const float*)d_in[b + 5];
      L[li].lin_v = (const float*)d_in[b + 6];
      for (int i = 0; i < 5; ++i) L[li].th[i] = (const float*)d_in[b + 7 + i];
    } else {
      L[li].lin_s = (const float*)d_in[b + 0];
      L[li].lin_v = (const float*)d_in[b + 1];
      // internal order [ss, vv, sv, vs, xx] <- sorted slots {0,3,1,2,4}
      const int map[5] = {0, 3, 1, 2, 4};
      for (int i = 0; i < 5; ++i) L[li].th[i] = (const float*)d_in[b + 2 + map[i]];
      for (int i = 0; i < 5; ++i) L[li].tm[i] = (const float*)d_in[b + 7 + map[i]];
    }
  }
  const float* head_w = (const float*)d_in[30];
  const float* head_b = (const float*)d_in[31];

  float* ws = (float*)d_ws;
  float* x        = ws; ws += (size_t)N_NODES * XDIM;
  float* node_msg = ws; ws += (size_t)N_NODES * MSGD;
  float* haggr    = ws; ws += (size_t)N_HEAVY * XDIM;
  float* hcnt     = ws; ws += (size_t)N_HEAVY;
  float* xtp      = ws; ws += (size_t)N_HEAVY * XDIM;

  embed_kernel<<<(N_NODES * 48 + 255) / 256, 256, 0, stream>>>(node_idx, embed_w, x);

  const int etiles = (N_EDGES + 31) / 32;
  const int eblocks = (etiles + 3) / 4;
  const int htiles = (N_HEAVY + 15) / 16;

  for (int li = 0; li < 2; ++li) {
    hipMemsetAsync(node_msg, 0, (size_t)N_NODES * MSGD * sizeof(float), stream);
    hipMemsetAsync(haggr, 0, (size_t)N_HEAVY * XDIM * sizeof(float), stream);
    hipMemsetAsync(hcnt, 0, (size_t)N_HEAVY * sizeof(float), stream);

    if (li == 0)
      edge_tp_kernel<24, 8><<<eblocks, 128, 0, stream>>>(
          x, src, dst, edge_attr, L[0].tm[0], L[0].tm[1], L[0].tm[2],
          L[0].tm[3], L[0].tm[4], node_msg);
    else
      edge_tp_kernel<16, 16><<<eblocks, 128, 0, stream>>>(
          x, src, dst, edge_attr, L[1].tm[0], L[1].tm[1], L[1].tm[2],
          L[1].tm[3], L[1].tm[4], node_msg);

    gate_kernel<<<(N_NODES + 127) / 128, 128, 0, stream>>>(
        node_msg, L[li].lin_s, L[li].lin_v, z, canonical, x, haggr, hcnt);

    heavy_tp_s_kernel<<<(htiles + 3) / 4, 128, 0, stream>>>(
        haggr, hcnt, L[li].th[0], L[li].th[1], xtp);
    heavy_tp_v_kernel<<<(htiles + 1) / 2, 64, 0, stream>>>(
        haggr, hcnt, L[li].th[2], L[li].th[3], L[li].th[4], xtp);

    merge_kernel<<<(N_NODES * XDIM + 255) / 256, 256, 0, stream>>>(
        xtp, z, canonical, x);
  }

  head_kernel<<<(N_NODES + 255) / 256, 256, 0, stream>>>(
      x, head_w, head_b, (float*)d_out);
}